// TensplitGCNLARGE_35519379538691
// MI455X (gfx1250) — compile-verified
//
#include <hip/hip_runtime.h>

typedef float v2f __attribute__((ext_vector_type(2)));
typedef float v8f __attribute__((ext_vector_type(8)));

#define IN_DIM 256
#define HID    256
#define OUTD   64
#define TM     32   // rows per workgroup

// ---------------------------------------------------------------------------
// Fused  H0 = relu(X @ W1) @ W2  for a 32-row tile, using V_WMMA_F32_16X16X4_F32.
// X tile and the ReLU'd intermediate are staged K-major in LDS so that the
// WMMA A-operand reads (lanes 0-15 = 16 consecutive rows at one K) are
// bank-conflict free. 8 waves / 256 threads per workgroup. LDS = 64 KB.
// ---------------------------------------------------------------------------
__global__ __launch_bounds__(256)
void gemm_fused_kernel(const float* __restrict__ X, const float* __restrict__ W1,
                       const float* __restrict__ W2, float* __restrict__ H0, int N)
{
    __shared__ float sX[IN_DIM][TM];   // transposed: sX[k][row]
    __shared__ float sM[HID][TM];      // transposed: sM[hid][row]

    const int t    = threadIdx.x;
    const int wave = t >> 5;
    const int lane = t & 31;
    const int hi   = lane >> 4;        // half-wave select
    const int r    = lane & 15;        // row (A) / col (B) within 16-tile
    const int kb   = hi * 2;           // K sub-offset per WMMA f32 16x16x4 layout
    const int row0 = blockIdx.x * TM;

    // ---- stage X tile into LDS (transposed), rows clamped at N-1 ----
    for (int i = 0; i < 8; ++i) {
        int f    = t + i * 256;        // float4 index; 64 float4 per row
        int rr   = f >> 6;             // 0..31
        int kq   = (f & 63) << 2;      // 0..252
        int grow = row0 + rr; if (grow >= N) grow = N - 1;
        const float4 g = reinterpret_cast<const float4*>(X + (size_t)grow * IN_DIM + kq)[0];
        sX[kq + 0][rr] = g.x; sX[kq + 1][rr] = g.y;
        sX[kq + 2][rr] = g.z; sX[kq + 3][rr] = g.w;
    }
    __syncthreads();

    // ---- stage 1: M = relu(Xtile @ W1). wave w owns col-blocks {2w,2w+1}, rows {0,1} ----
    {
        const int cb0 = wave * 2;
        const int c0  = cb0 * 16 + r;
        const int c1  = c0 + 16;
        v8f a00 = {}, a01 = {}, a10 = {}, a11 = {};
        for (int k = 0; k < IN_DIM; k += 4) {
            v2f a0, a1, b0, b1;
            a0.x = sX[k + kb][r];          a0.y = sX[k + kb + 1][r];
            a1.x = sX[k + kb][16 + r];     a1.y = sX[k + kb + 1][16 + r];
            b0.x = W1[(k + kb) * HID + c0]; b0.y = W1[(k + kb + 1) * HID + c0];
            b1.x = W1[(k + kb) * HID + c1]; b1.y = W1[(k + kb + 1) * HID + c1];
            a00 = __builtin_amdgcn_wmma_f32_16x16x4_f32(false, a0, false, b0, (short)0, a00, false, false);
            a01 = __builtin_amdgcn_wmma_f32_16x16x4_f32(false, a0, false, b1, (short)0, a01, false, false);
            a10 = __builtin_amdgcn_wmma_f32_16x16x4_f32(false, a1, false, b0, (short)0, a10, false, false);
            a11 = __builtin_amdgcn_wmma_f32_16x16x4_f32(false, a1, false, b1, (short)0, a11, false, false);
        }
        // ReLU + store D (lane holds rows v+8*hi of col c) transposed into sM
        for (int v = 0; v < 8; ++v) {
            const int rl = v + 8 * hi;
            float x;
            x = a00[v]; sM[c0][rl]      = x > 0.f ? x : 0.f;
            x = a01[v]; sM[c1][rl]      = x > 0.f ? x : 0.f;
            x = a10[v]; sM[c0][16 + rl] = x > 0.f ? x : 0.f;
            x = a11[v]; sM[c1][16 + rl] = x > 0.f ? x : 0.f;
        }
    }
    __syncthreads();

    // ---- stage 2: out = M @ W2. wave w owns tile (rb = w>>2, cb = w&3) ----
    {
        const int rb = wave >> 2, cb = wave & 3;
        const int c  = cb * 16 + r;
        v8f acc = {};
        for (int k = 0; k < HID; k += 4) {
            v2f a, b;
            a.x = sM[k + kb][rb * 16 + r]; a.y = sM[k + kb + 1][rb * 16 + r];
            b.x = W2[(k + kb) * OUTD + c]; b.y = W2[(k + kb + 1) * OUTD + c];
            acc = __builtin_amdgcn_wmma_f32_16x16x4_f32(false, a, false, b, (short)0, acc, false, false);
        }
        for (int v = 0; v < 8; ++v) {
            const int grow = row0 + rb * 16 + v + 8 * hi;
            if (grow < N) H0[(size_t)grow * OUTD + c] = acc[v];
        }
    }
}

// ---------------------------------------------------------------------------
// Edge-list SpMM: Hout[row] += val * Hin[col]. 16 threads per edge, float4
// gather (L2-resident rows), 4 hardware fp32 atomics to the output row.
// ---------------------------------------------------------------------------
__global__ __launch_bounds__(256)
void spmm_edges_kernel(const float* __restrict__ vals, const int* __restrict__ rows,
                       const int* __restrict__ cols, const float* __restrict__ Hin,
                       float* __restrict__ Hout, int E)
{
    long long tid = (long long)blockIdx.x * blockDim.x + threadIdx.x;
    if (tid >= (long long)E * 16) return;
    const int e  = (int)(tid >> 4);
    const int c4 = (int)(tid & 15);
    const float v  = vals[e];
    const int col  = cols[e];
    const int row  = rows[e];
    const float4 x = reinterpret_cast<const float4*>(Hin)[(size_t)col * 16 + c4];
    float* base = Hout + (size_t)row * OUTD + c4 * 4;
    unsafeAtomicAdd(base + 0, v * x.x);
    unsafeAtomicAdd(base + 1, v * x.y);
    unsafeAtomicAdd(base + 2, v * x.z);
    unsafeAtomicAdd(base + 3, v * x.w);
}

__global__ __launch_bounds__(256)
void zero_f4_kernel(float4* __restrict__ p, int n4)
{
    int i = blockIdx.x * blockDim.x + threadIdx.x;
    if (i < n4) p[i] = make_float4(0.f, 0.f, 0.f, 0.f);
}

extern "C" void kernel_launch(void* const* d_in, const int* in_sizes, int n_in,
                              void* d_out, int out_size, void* d_ws, size_t ws_size,
                              hipStream_t stream)
{
    const float* X    = (const float*)d_in[0];
    const float* W1   = (const float*)d_in[1];
    const float* W2   = (const float*)d_in[2];
    const float* vals = (const float*)d_in[3];
    const int*   rows = (const int*)d_in[4];
    const int*   cols = (const int*)d_in[5];
    float* out = (float*)d_out;

    const int N = in_sizes[0] / IN_DIM;
    const int E = in_sizes[3];

    const size_t hbytes = (size_t)N * OUTD * sizeof(float);
    const int n4 = (N * OUTD) / 4;
    const int zblocks = (n4 + 255) / 256;
    const int gblocks = (N + TM - 1) / TM;
    const int sblocks = (int)(((long long)E * 16 + 255) / 256);

    float* h0 = (float*)d_ws;

    if (ws_size >= 2 * hbytes) {
        // ping-pong entirely in workspace, final layer lands in d_out
        float* h1 = h0 + (size_t)N * OUTD;
        gemm_fused_kernel<<<gblocks, 256, 0, stream>>>(X, W1, W2, h0, N);
        zero_f4_kernel<<<zblocks, 256, 0, stream>>>((float4*)h1, n4);
        spmm_edges_kernel<<<sblocks, 256, 0, stream>>>(vals, rows, cols, h0, h1, E);
        zero_f4_kernel<<<zblocks, 256, 0, stream>>>((float4*)out, n4);
        spmm_edges_kernel<<<sblocks, 256, 0, stream>>>(vals, rows, cols, h1, out, E);
    } else {
        // small workspace: use d_out as the second buffer, d2d copy at the end
        gemm_fused_kernel<<<gblocks, 256, 0, stream>>>(X, W1, W2, h0, N);
        zero_f4_kernel<<<zblocks, 256, 0, stream>>>((float4*)out, n4);
        spmm_edges_kernel<<<sblocks, 256, 0, stream>>>(vals, rows, cols, h0, out, E);
        zero_f4_kernel<<<zblocks, 256, 0, stream>>>((float4*)h0, n4);
        spmm_edges_kernel<<<sblocks, 256, 0, stream>>>(vals, rows, cols, out, h0, E);
        hipMemcpyAsync(out, h0, hbytes, hipMemcpyDeviceToDevice, stream);
    }
}